// KGAT_84722524881132
// MI455X (gfx1250) — compile-verified
//
#include <hip/hip_runtime.h>

typedef float v2f __attribute__((ext_vector_type(2)));
typedef float v8f __attribute__((ext_vector_type(8)));

#define NEG_SLOPE 0.01f

__device__ __forceinline__ float leaky(float x) {
    return x >= 0.f ? x : NEG_SLOPE * x;
}

// ---------------------------------------------------------------- zero
__global__ void zero_f32(float* __restrict__ p, long n) {
    long i = (long)blockIdx.x * blockDim.x + threadIdx.x;
    if (i < n) p[i] = 0.f;
}

// ---------------------------------------------------------------- SpMM (COO scatter-atomic)
// y[row] += val * x[col]; one thread per edge, float4 gathers, f32 atomics (L2-resident).
template<int D>
__global__ void spmm_atomic(const int* __restrict__ erow, const int* __restrict__ ecol,
                            const float* __restrict__ evalv, const float* __restrict__ x,
                            float* __restrict__ y, int E) {
    int e = blockIdx.x * blockDim.x + threadIdx.x;
    if (e >= E) return;
    int r = erow[e];
    int c = ecol[e];
    float v = evalv[e];
    const float4* xr = (const float4*)(x + (long)c * D);
    float* yr = y + (long)r * D;
#pragma unroll
    for (int g = 0; g < D / 4; ++g) {
        float4 xv = xr[g];
        atomicAdd(&yr[4 * g + 0], v * xv.x);
        atomicAdd(&yr[4 * g + 1], v * xv.y);
        atomicAdd(&yr[4 * g + 2], v * xv.z);
        atomicAdd(&yr[4 * g + 3], v * xv.w);
    }
}

// ---------------------------------------------------------------- bi-interaction layer 0
// ego,side: [N,64]; W1,W2: [64,32]; ego1: [N,32]
// One wave32 per 16-row tile. K=64 as 16 WMMA steps of K=4; N=32 as two 16-col tiles.
// A-frag (16x4 f32): lane<16 -> m=lane, {K=ks,K=ks+1}; lane>=16 -> {K=ks+2,K=ks+3} (ks folded below)
// B-frag (4x16 f32): lane%16 = n, VGPR pair holds rows {ks, ks+1}
// D (16x16 f32): lane%16 = n, VGPR v -> m = v + 8*(lane/16)
__global__ void bi_layer0(const float* __restrict__ ego, const float* __restrict__ side,
                          const float* __restrict__ W1, const float* __restrict__ b1,
                          const float* __restrict__ W2, const float* __restrict__ b2,
                          float* __restrict__ ego1, int N) {
    int wave = (int)((blockIdx.x * blockDim.x + threadIdx.x) >> 5);
    int lane = threadIdx.x & 31;
    int m0 = wave * 16;
    if (m0 >= N) return;                 // wave-uniform: EXEC stays all-ones for live waves
    int hh = lane >> 4;                  // lane half (selects K pair / M offset 8)
    int lm = lane & 15;                  // A: m index, B/D: n index

    v8f cs0 = {}, cs1 = {}, cp0 = {}, cp1 = {};

#pragma unroll 4
    for (int kk = 0; kk < 16; ++kk) {
        int ks = kk * 4 + 2 * hh;
        int row = m0 + lm;
        float2 e2 = *(const float2*)(ego  + (long)row * 64 + ks);
        float2 s2 = *(const float2*)(side + (long)row * 64 + ks);
        v2f as = { e2.x + s2.x, e2.y + s2.y };   // (ego + side) fragment
        v2f ap = { e2.x * s2.x, e2.y * s2.y };   // (ego * side) fragment
        v2f b1a = { W1[ks * 32 + lm],      W1[(ks + 1) * 32 + lm] };
        v2f b1b = { W1[ks * 32 + 16 + lm], W1[(ks + 1) * 32 + 16 + lm] };
        v2f b2a = { W2[ks * 32 + lm],      W2[(ks + 1) * 32 + lm] };
        v2f b2b = { W2[ks * 32 + 16 + lm], W2[(ks + 1) * 32 + 16 + lm] };
        cs0 = __builtin_amdgcn_wmma_f32_16x16x4_f32(false, as, false, b1a, (short)0, cs0, false, false);
        cs1 = __builtin_amdgcn_wmma_f32_16x16x4_f32(false, as, false, b1b, (short)0, cs1, false, false);
        cp0 = __builtin_amdgcn_wmma_f32_16x16x4_f32(false, ap, false, b2a, (short)0, cp0, false, false);
        cp1 = __builtin_amdgcn_wmma_f32_16x16x4_f32(false, ap, false, b2b, (short)0, cp1, false, false);
    }

    float bb1a = b1[lm], bb1b = b1[16 + lm];
    float bb2a = b2[lm], bb2b = b2[16 + lm];
#pragma unroll
    for (int v = 0; v < 8; ++v) {
        int m = m0 + v + 8 * hh;
        float r0 = leaky(cs0[v] + bb1a) + leaky(cp0[v] + bb2a);
        float r1 = leaky(cs1[v] + bb1b) + leaky(cp1[v] + bb2b);
        ego1[(long)m * 32 + lm]      = r0;
        ego1[(long)m * 32 + 16 + lm] = r1;
    }
}

// ---------------------------------------------------------------- bi-interaction layer 1
// ego,side: [N,32]; W1,W2: [32,16]; writes unnormalized ego2 to out[:, 96:112]
__global__ void bi_layer1(const float* __restrict__ ego, const float* __restrict__ side,
                          const float* __restrict__ W1, const float* __restrict__ b1,
                          const float* __restrict__ W2, const float* __restrict__ b2,
                          float* __restrict__ out, int N) {
    int wave = (int)((blockIdx.x * blockDim.x + threadIdx.x) >> 5);
    int lane = threadIdx.x & 31;
    int m0 = wave * 16;
    if (m0 >= N) return;
    int hh = lane >> 4;
    int lm = lane & 15;

    v8f cs = {}, cp = {};

#pragma unroll
    for (int kk = 0; kk < 8; ++kk) {
        int ks = kk * 4 + 2 * hh;
        int row = m0 + lm;
        float2 e2 = *(const float2*)(ego  + (long)row * 32 + ks);
        float2 s2 = *(const float2*)(side + (long)row * 32 + ks);
        v2f as = { e2.x + s2.x, e2.y + s2.y };
        v2f ap = { e2.x * s2.x, e2.y * s2.y };
        v2f bb1 = { W1[ks * 16 + lm], W1[(ks + 1) * 16 + lm] };
        v2f bb2 = { W2[ks * 16 + lm], W2[(ks + 1) * 16 + lm] };
        cs = __builtin_amdgcn_wmma_f32_16x16x4_f32(false, as, false, bb1, (short)0, cs, false, false);
        cp = __builtin_amdgcn_wmma_f32_16x16x4_f32(false, ap, false, bb2, (short)0, cp, false, false);
    }

    float bia = b1[lm], bib = b2[lm];
#pragma unroll
    for (int v = 0; v < 8; ++v) {
        int m = m0 + v + 8 * hh;
        out[(long)m * 112 + 96 + lm] = leaky(cs[v] + bia) + leaky(cp[v] + bib);
    }
}

// ---------------------------------------------------------------- row L2-normalize (ws -> out cols)
template<int D>
__global__ void norm_rows_to_out(const float* __restrict__ src, float* __restrict__ out,
                                 int N, int col0) {
    int r = blockIdx.x * blockDim.x + threadIdx.x;
    if (r >= N) return;
    const float* p = src + (long)r * D;
    float ss = 0.f;
#pragma unroll
    for (int j = 0; j < D; ++j) { float v = p[j]; ss += v * v; }
    float inv = 1.f / fmaxf(sqrtf(ss), 1e-12f);
    float* o = out + (long)r * 112 + col0;
#pragma unroll
    for (int j = 0; j < D; ++j) o[j] = p[j] * inv;
}

// ---------------------------------------------------------------- in-place normalize of out[:, 96:112]
__global__ void norm_inplace16(float* __restrict__ out, int N) {
    int r = blockIdx.x * blockDim.x + threadIdx.x;
    if (r >= N) return;
    float* p = out + (long)r * 112 + 96;
    float ss = 0.f;
#pragma unroll
    for (int j = 0; j < 16; ++j) ss += p[j] * p[j];
    float inv = 1.f / fmaxf(sqrtf(ss), 1e-12f);
#pragma unroll
    for (int j = 0; j < 16; ++j) p[j] *= inv;
}

// ---------------------------------------------------------------- out[:, 0:64] = node_embed
__global__ void copy_embed(const float* __restrict__ x, float* __restrict__ out, long n) {
    long i = (long)blockIdx.x * blockDim.x + threadIdx.x;
    if (i >= n) return;
    long r = i >> 6;       // /64
    int  c = (int)(i & 63);
    out[r * 112 + c] = x[i];
}

// ================================================================ launch
extern "C" void kernel_launch(void* const* d_in, const int* in_sizes, int n_in,
                              void* d_out, int out_size, void* d_ws, size_t ws_size,
                              hipStream_t stream) {
    const float* node_embed = (const float*)d_in[0];
    const int*   erow  = (const int*)d_in[1];
    const int*   ecol  = (const int*)d_in[2];
    const float* evalv = (const float*)d_in[3];
    const float* W1_0 = (const float*)d_in[4];
    const float* b1_0 = (const float*)d_in[5];
    const float* W2_0 = (const float*)d_in[6];
    const float* b2_0 = (const float*)d_in[7];
    const float* W1_1 = (const float*)d_in[8];
    const float* b1_1 = (const float*)d_in[9];
    const float* W2_1 = (const float*)d_in[10];
    const float* b2_1 = (const float*)d_in[11];

    const int N = in_sizes[0] / 64;
    const int E = in_sizes[1];
    float* out  = (float*)d_out;
    float* side = (float*)d_ws;                 // N*64 floats; reused as side1 (N*32)
    float* ego1 = side + (long)N * 64;          // N*32 floats

    const int B = 256;
    const int tiles  = (N + 15) / 16;           // one wave32 per 16-row tile
    const int biBlks = (tiles + 7) / 8;         // 8 waves per 256-thread block

    // ---- layer 0 ----
    zero_f32<<<(int)(((long)N * 64 + B - 1) / B), B, 0, stream>>>(side, (long)N * 64);
    spmm_atomic<64><<<(E + B - 1) / B, B, 0, stream>>>(erow, ecol, evalv, node_embed, side, E);
    bi_layer0<<<biBlks, B, 0, stream>>>(node_embed, side, W1_0, b1_0, W2_0, b2_0, ego1, N);
    norm_rows_to_out<32><<<(N + B - 1) / B, B, 0, stream>>>(ego1, out, N, 64);
    copy_embed<<<(int)(((long)N * 64 + B - 1) / B), B, 0, stream>>>(node_embed, out, (long)N * 64);

    // ---- layer 1 (side buffer reused) ----
    zero_f32<<<(int)(((long)N * 32 + B - 1) / B), B, 0, stream>>>(side, (long)N * 32);
    spmm_atomic<32><<<(E + B - 1) / B, B, 0, stream>>>(erow, ecol, evalv, ego1, side, E);
    bi_layer1<<<biBlks, B, 0, stream>>>(ego1, side, W1_1, b1_1, W2_1, b2_1, out, N);
    norm_inplace16<<<(N + B - 1) / B, B, 0, stream>>>(out, N);
}